// GAT_17575006175718
// MI455X (gfx1250) — compile-verified
//
#include <hip/hip_runtime.h>

typedef __attribute__((ext_vector_type(2))) float v2f;
typedef __attribute__((ext_vector_type(8))) float v8f;

// ---------------------------------------------------------------------------
// GEMM: C[N,OC] = A[N,IC] @ B[IC,OC], f32, V_WMMA_F32_16X16X4_F32.
// One wave per 16x16 tile; K stepped by 4.
// A 16x4 layout: lanes 0-15 rows M=0..15 K={k,k+1}; lanes 16-31 K={k+2,k+3}.
// B 4x16 layout mirrors (row striped across lanes within VGPR).
// D: VGPR r, lanes0-15 -> (M=r, N=l); lanes16-31 -> (M=r+8, N=l).
// ---------------------------------------------------------------------------
__global__ void gat_gemm_wmma(const float* __restrict__ A, const float* __restrict__ B,
                              float* __restrict__ C, int IC, int OC, int CT, int nTiles) {
    int wid = (blockIdx.x * blockDim.x + threadIdx.x) >> 5;
    if (wid >= nTiles) return;                 // wave-uniform exit: EXEC all-1s for WMMA
    int lane = threadIdx.x & 31;
    int rt = wid / CT, ct = wid - rt * CT;
    int m0 = rt << 4, n0 = ct << 4;
    int half = lane >> 4, l = lane & 15;

    const float* arow = A + (size_t)(m0 + l) * IC;
    const float* bcol = B + n0 + l;
    v8f acc = {};
    for (int k = 0; k < IC; k += 4) {
        int ka = k + (half << 1);
        v2f a = *(const v2f*)(arow + ka);       // 8B aligned: ka even, rows 128 floats
        v2f b;
        b.x = bcol[(size_t)ka * OC];
        b.y = bcol[(size_t)(ka + 1) * OC];
        acc = __builtin_amdgcn_wmma_f32_16x16x4_f32(
            /*neg_a=*/false, a, /*neg_b=*/false, b,
            /*c_mod=*/(short)0, acc, /*reuse_a=*/false, /*reuse_b=*/false);
    }
    float* crow = C + (size_t)(m0 + half * 8) * OC + n0 + l;
#pragma unroll
    for (int r = 0; r < 8; ++r) crow[(size_t)r * OC] = acc[r];
}

// ---------------------------------------------------------------------------
// Per-(node,head) attention logit pieces + init of segment max / denom.
// ---------------------------------------------------------------------------
__global__ void gat_scores(const float* __restrict__ x, const float* __restrict__ as,
                           const float* __restrict__ ad, float* __restrict__ es,
                           float* __restrict__ ed, float* __restrict__ mx,
                           float* __restrict__ dn, int N, int H, int OC) {
    int idx = blockIdx.x * blockDim.x + threadIdx.x;
    if (idx >= N * H) return;
    int n = idx / H, h = idx - n * H;
    const float* xr = x + (size_t)n * OC + h * 32;
    const float* ar = as + h * 32;
    const float* dr = ad + h * 32;
    float s0 = 0.f, s1 = 0.f;
#pragma unroll
    for (int c = 0; c < 32; ++c) { float v = xr[c]; s0 += v * ar[c]; s1 += v * dr[c]; }
    es[idx] = s0;
    ed[idx] = s1;
    mx[idx] = -3.0e38f;
    dn[idx] = 0.f;
}

__device__ __forceinline__ void edge_sd(const int* __restrict__ ei, int e, int E,
                                        int& s, int& d) {
    if (e < E) { s = ei[e]; d = ei[E + e]; } else { s = d = e - E; }  // self loops
}

__device__ __forceinline__ float leaky(float v) { return v > 0.f ? v : 0.2f * v; }

// order-preserving float atomic max via sign-split int/uint atomics
__device__ __forceinline__ void atomicMaxF(float* addr, float v) {
    if (v >= 0.f) atomicMax((int*)addr, __float_as_int(v));
    else          atomicMin((unsigned int*)addr, __float_as_uint(v));
}

__global__ void gat_edge_max(const int* __restrict__ ei, const float* __restrict__ es,
                             const float* __restrict__ ed, float* __restrict__ mx,
                             int E, int Etot, int H) {
    int e = blockIdx.x * blockDim.x + threadIdx.x;
    if (e >= Etot) return;
    int s, d; edge_sd(ei, e, E, s, d);
    for (int h = 0; h < H; ++h) {
        float v = leaky(es[s * H + h] + ed[d * H + h]);
        atomicMaxF(&mx[d * H + h], v);
    }
}

__global__ void gat_edge_den(const int* __restrict__ ei, const float* __restrict__ es,
                             const float* __restrict__ ed, const float* __restrict__ mx,
                             float* __restrict__ dn, int E, int Etot, int H) {
    int e = blockIdx.x * blockDim.x + threadIdx.x;
    if (e >= Etot) return;
    int s, d; edge_sd(ei, e, E, s, d);
    for (int h = 0; h < H; ++h) {
        float v = leaky(es[s * H + h] + ed[d * H + h]);
        atomicAdd(&dn[d * H + h], expf(v - mx[d * H + h]));
    }
}

// One wave per edge; lanes 0..H-1 compute per-head alpha, shfl-broadcast,
// then 32 lanes stream OC channels of gather*alpha -> atomic add.
__global__ void gat_edge_aggr(const int* __restrict__ ei, const float* __restrict__ es,
                              const float* __restrict__ ed, const float* __restrict__ mx,
                              const float* __restrict__ dn, const float* __restrict__ x,
                              float* __restrict__ out, int E, int Etot, int H, int OC) {
    int e = (blockIdx.x * blockDim.x + threadIdx.x) >> 5;
    if (e >= Etot) return;
    int lane = threadIdx.x & 31;
    int s, d; edge_sd(ei, e, E, s, d);
    float alpha = 0.f;
    if (lane < H) {
        float v = leaky(es[s * H + lane] + ed[d * H + lane]);
        alpha = expf(v - mx[d * H + lane]) / (dn[d * H + lane] + 1e-16f);
    }
    const float* xs = x + (size_t)s * OC;
    float* od = out + (size_t)d * OC;
    for (int c = lane; c < OC; c += 32) {
        float a = __shfl(alpha, c >> 5, 32);   // head = channel/32, wave-uniform per iter
        atomicAdd(&od[c], xs[c] * a);
    }
}

// ---------------------------------------------------------------------------
// BN: per-channel sum/sumsq (LDS accumulate -> global atomics), finalize, apply+ELU.
// (GAT bias cancels inside BN: x+b - mean(x+b) == x - mean(x).)
// ---------------------------------------------------------------------------
__global__ void gat_stats(const float* __restrict__ a, float* __restrict__ st,
                          int total, int OC) {
    __shared__ float sh[256];
    for (int i = threadIdx.x; i < 2 * OC; i += blockDim.x) sh[i] = 0.f;
    __syncthreads();
    int ocm = OC - 1;
    for (int idx = blockIdx.x * blockDim.x + threadIdx.x; idx < total;
         idx += gridDim.x * blockDim.x) {
        float v = a[idx];
        int c = idx & ocm;
        atomicAdd(&sh[c], v);
        atomicAdd(&sh[OC + c], v * v);
    }
    __syncthreads();
    for (int i = threadIdx.x; i < 2 * OC; i += blockDim.x) atomicAdd(&st[i], sh[i]);
}

__global__ void gat_bn_finalize(float* __restrict__ st, const float* __restrict__ g,
                                const float* __restrict__ be, float invN, int OC) {
    int c = threadIdx.x;
    if (c >= OC) return;
    float mean = st[c] * invN;
    float var  = st[OC + c] * invN - mean * mean;
    float sc   = g[c] * rsqrtf(var + 1e-5f);
    st[256 + c] = sc;
    st[384 + c] = be[c] - mean * sc;
}

__global__ void gat_bn_apply(float* __restrict__ a, const float* __restrict__ st,
                             int total, int OC) {
    int idx = blockIdx.x * blockDim.x + threadIdx.x;
    if (idx >= total) return;
    int c = idx & (OC - 1);
    float v = st[256 + c] * a[idx] + st[384 + c];
    a[idx] = v > 0.f ? v : expm1f(v);          // ELU
}

// ---------------------------------------------------------------------------
// Classifier head: [N,32] -> elu(@cw1+cb1) -> @cw2+cb2 -> [N,2]
// ---------------------------------------------------------------------------
__global__ void gat_mlp(const float* __restrict__ h, const float* __restrict__ cw1,
                        const float* __restrict__ cb1, const float* __restrict__ cw2,
                        const float* __restrict__ cb2, float* __restrict__ out, int N) {
    int n = blockIdx.x * blockDim.x + threadIdx.x;
    if (n >= N) return;
    float hv[32];
#pragma unroll
    for (int i = 0; i < 32; ++i) hv[i] = h[(size_t)n * 32 + i];
    float o0 = cb2[0], o1 = cb2[1];
#pragma unroll
    for (int j = 0; j < 16; ++j) {
        float t = cb1[j];
#pragma unroll
        for (int i = 0; i < 32; ++i) t += hv[i] * cw1[i * 16 + j];
        t = t > 0.f ? t : expm1f(t);
        o0 += t * cw2[j * 2 + 0];
        o1 += t * cw2[j * 2 + 1];
    }
    out[(size_t)n * 2 + 0] = o0;
    out[(size_t)n * 2 + 1] = o1;
}

// ---------------------------------------------------------------------------
extern "C" void kernel_launch(void* const* d_in, const int* in_sizes, int n_in,
                              void* d_out, int out_size, void* d_ws, size_t ws_size,
                              hipStream_t stream) {
    const int N = 50000, E = 1600000, Etot = E + N;
    const float* x  = (const float*)d_in[0];
    const int*   ei = (const int*)d_in[1];

    // workspace layout (floats): ~54.5 MB
    float* xb = (float*)d_ws;               // [N,128] post-GEMM features
    float* hb = xb + (size_t)N * 128;       // [N,128] aggregation out / next-layer in
    float* es = hb + (size_t)N * 128;       // [N,4]
    float* ed = es + (size_t)N * 4;         // [N,4]
    float* mx = ed + (size_t)N * 4;         // [N,4]
    float* dn = mx + (size_t)N * 4;         // [N,4]
    float* st = dn + (size_t)N * 4;         // 512: sum|sumsq|scale|shift

    const float* Ws[3] = {(const float*)d_in[2],  (const float*)d_in[8],  (const float*)d_in[14]};
    const float* As[3] = {(const float*)d_in[3],  (const float*)d_in[9],  (const float*)d_in[15]};
    const float* Ad[3] = {(const float*)d_in[4],  (const float*)d_in[10], (const float*)d_in[16]};
    const float* Gm[3] = {(const float*)d_in[6],  (const float*)d_in[12], (const float*)d_in[18]};
    const float* Bt[3] = {(const float*)d_in[7],  (const float*)d_in[13], (const float*)d_in[19]};
    const int Hh[3]  = {4, 4, 1};
    const int OCs[3] = {128, 128, 32};

    const float* hin = x;
    for (int l = 0; l < 3; ++l) {
        const int OC = OCs[l], H = Hh[l], IC = 128;
        const int CT = OC / 16;
        const int nTiles = (N / 16) * CT;

        gat_gemm_wmma<<<(nTiles + 7) / 8, 256, 0, stream>>>(hin, Ws[l], xb, IC, OC, CT, nTiles);
        gat_scores<<<(N * H + 255) / 256, 256, 0, stream>>>(xb, As[l], Ad[l], es, ed, mx, dn, N, H, OC);
        hipMemsetAsync(hb, 0, (size_t)N * OC * sizeof(float), stream);
        hipMemsetAsync(st, 0, 256 * sizeof(float), stream);
        gat_edge_max<<<(Etot + 255) / 256, 256, 0, stream>>>(ei, es, ed, mx, E, Etot, H);
        gat_edge_den<<<(Etot + 255) / 256, 256, 0, stream>>>(ei, es, ed, mx, dn, E, Etot, H);
        gat_edge_aggr<<<(Etot + 7) / 8, 256, 0, stream>>>(ei, es, ed, mx, dn, xb, hb, E, Etot, H, OC);
        gat_stats<<<512, 256, 0, stream>>>(hb, st, N * OC, OC);
        gat_bn_finalize<<<1, 128, 0, stream>>>(st, Gm[l], Bt[l], 1.0f / (float)N, OC);
        gat_bn_apply<<<(N * OC + 255) / 256, 256, 0, stream>>>(hb, st, N * OC, OC);
        hin = hb;
    }

    gat_mlp<<<(N + 255) / 256, 256, 0, stream>>>(hb, (const float*)d_in[20], (const float*)d_in[21],
                                                 (const float*)d_in[22], (const float*)d_in[23],
                                                 (float*)d_out, N);
    (void)in_sizes; (void)n_in; (void)out_size; (void)ws_size;
}